// SoftRasterizer_35278861369365
// MI455X (gfx1250) — compile-verified
//
#include <hip/hip_runtime.h>

// Soft rasterizer as 32 on-the-fly-operand GEMMs on the f32 WMMA pipe.
// out[bk][h][w] = sum_n exp(-(h-y_n)^2 c)/z_n * exp(-(w-x_n)^2 c)

#define H_DIM  256
#define W_DIM  256
#define NPTS   4096
#define NIMG   32          // B*K = 4*8
#define TILE   64          // output tile edge
#define NC     32          // points per chunk (K-block)
#define THREADS 128        // 4 waves
#define APAD   80          // LDS row stride: 2*80 % 64 == 32 -> conflict-free half-wave split

typedef __attribute__((ext_vector_type(2))) float v2f;
typedef __attribute__((ext_vector_type(8))) float v8f;

__global__ __launch_bounds__(THREADS)
void SoftRasterizer_wmma_kernel(const float* __restrict__ pts2d,  // (NIMG, NPTS, 2)
                                const float* __restrict__ ptsz,   // (NIMG, NPTS)
                                float* __restrict__ out)          // (NIMG, H, W)
{
    __shared__ float sX[NPTS];
    __shared__ float sY[NPTS];
    __shared__ float sIZ[NPTS];
    __shared__ float sA[NC][APAD];   // wy tile: [k][h]
    __shared__ float sB[NC][APAD];   // wx tile: [k][w]

    const int tid  = threadIdx.x;
    const int lane = tid & 31;
    const int wave = tid >> 5;          // 0..3 -> 16-row strip of the 64x64 tile
    const int bk   = blockIdx.y;        // image index (B*K)
    const int h0   = (blockIdx.x >> 2) * TILE;
    const int w0   = (blockIdx.x & 3)  * TILE;

    const float IV = 1.0f / (2.0f * (1.5f * 1.5f + 1e-8f));  // INV_VAR

    // ---- preload all point data for this image into LDS (48 KB) ----
    const float2* p2 = (const float2*)(pts2d + (size_t)bk * NPTS * 2);
    const float*  pz = ptsz + (size_t)bk * NPTS;
    for (int i = tid; i < NPTS; i += THREADS) {
        float2 p = p2[i];
        sX[i]  = p.x;
        sY[i]  = p.y;
        sIZ[i] = 1.0f / fmaxf(pz[i], 1e-5f);
    }
    __syncthreads();

    v8f acc[4];
    #pragma unroll
    for (int w = 0; w < 4; ++w)
        #pragma unroll
        for (int j = 0; j < 8; ++j) acc[w][j] = 0.0f;

    const int hi   = lane >> 4;          // half-wave select (K split)
    const int l15  = lane & 15;
    const int hrow = wave * 16 + l15;    // A-matrix row (h within tile)

    for (int n0 = 0; n0 < NPTS; n0 += NC) {
        // ---- generate A (wy/z) and B (wx) chunk tiles into LDS ----
        for (int i = tid; i < NC * TILE; i += THREADS) {
            const int kc = i >> 6;               // / TILE
            const int e  = i & (TILE - 1);
            const float y  = sY[n0 + kc];
            const float x  = sX[n0 + kc];
            const float dy = (float)(h0 + e) - y;
            const float dx = (float)(w0 + e) - x;
            sA[kc][e] = __expf(-dy * dy * IV) * sIZ[n0 + kc];
            sB[kc][e] = __expf(-dx * dx * IV);
        }
        __syncthreads();

        // ---- consume: 8 K-steps of 4, 4 W-subtiles each ----
        #pragma unroll
        for (int ks = 0; ks < NC / 4; ++ks) {
            const int kb = ks * 4 + hi * 2;      // this half-wave's K pair
            v2f a;
            a.x = sA[kb][hrow];
            a.y = sA[kb + 1][hrow];
            #pragma unroll
            for (int w = 0; w < 4; ++w) {
                v2f b;
                b.x = sB[kb][w * 16 + l15];
                b.y = sB[kb + 1][w * 16 + l15];
                acc[w] = __builtin_amdgcn_wmma_f32_16x16x4_f32(
                    false, a, false, b, (short)0, acc[w], false, false);
            }
        }
        __syncthreads();
    }

    // ---- store: C/D layout = VGPR j holds rows M=j (lanes 0-15) / M=j+8 (lanes 16-31) ----
    float* o = out + ((size_t)bk * H_DIM + (h0 + wave * 16)) * W_DIM + w0;
    #pragma unroll
    for (int w = 0; w < 4; ++w) {
        #pragma unroll
        for (int j = 0; j < 8; ++j) {
            const int r = j + 8 * hi;
            o[(size_t)r * W_DIM + w * 16 + l15] = acc[w][j];
        }
    }
}

extern "C" void kernel_launch(void* const* d_in, const int* in_sizes, int n_in,
                              void* d_out, int out_size, void* d_ws, size_t ws_size,
                              hipStream_t stream) {
    const float* pts2d = (const float*)d_in[0];   // (4,8,4096,2) f32
    const float* ptsz  = (const float*)d_in[1];   // (4,8,4096)   f32
    float* out = (float*)d_out;                   // (4,8,256,256) f32

    dim3 grid(16, NIMG);   // 16 tiles of 64x64 per image, 32 images
    SoftRasterizer_wmma_kernel<<<grid, THREADS, 0, stream>>>(pts2d, ptsz, out);
}